// VectorQuantize_57492432224576
// MI455X (gfx1250) — compile-verified
//
#include <hip/hip_runtime.h>

typedef __attribute__((ext_vector_type(2))) float v2f;
typedef __attribute__((ext_vector_type(8))) float v8f;
typedef __attribute__((ext_vector_type(4))) float f4;

#define DDIM 256
#define KCB  512

// Kernel 1: 0.5 * ||cb_k||^2 for each codebook entry -> d_ws
__global__ void vq_norms_kernel(const float* __restrict__ cb,
                                float* __restrict__ normhalf) {
    int k = blockIdx.x * blockDim.x + threadIdx.x;
    if (k < KCB) {
        const float* row = cb + (size_t)k * DDIM;
        float s = 0.0f;
        for (int d = 0; d < DDIM; ++d) s += row[d] * row[d];
        normhalf[k] = 0.5f * s;
    }
}

// Kernel 2: per-wave 16-row strip; fp32 WMMA distance GEMM + argmin + gather.
__global__ __launch_bounds__(256) void vq_main_kernel(
    const float* __restrict__ x, const float* __restrict__ cb,
    const float* __restrict__ normhalf, float* __restrict__ out, int nrows) {
    const int lane  = threadIdx.x & 31;
    const int wave  = threadIdx.x >> 5;
    const int strip = blockIdx.x * 8 + wave;
    const int row0  = strip * 16;
    if (row0 >= nrows) return;  // wave-uniform; EXEC stays all-ones

    const int lm     = lane & 15;
    const int half   = lane >> 4;   // 0: K=0,1  1: K=2,3 (A/B fp32 WMMA layout)
    const int dphase = half * 2;
    const int myrow  = row0 + lm;

    // Register-resident A strip: za[t] = x[myrow, 4t+dphase .. 4t+dphase+1]
    v2f za[64];
    const float* xrow = x + (size_t)myrow * DDIM + dphase;
#pragma unroll
    for (int t = 0; t < 64; ++t)
        za[t] = *(const v2f*)(xrow + 4 * t);

    float bestv[8];
    int   besti[8];
#pragma unroll
    for (int j = 0; j < 8; ++j) { bestv[j] = -3.4e38f; besti[j] = 0; }

#pragma unroll 1
    for (int kt = 0; kt < KCB / 16; ++kt) {
        const int kcol = kt * 16 + lm;           // this lane's codebook column
        const float cinit = -normhalf[kcol];     // score = z.cb - 0.5||cb||^2
        v8f c;
#pragma unroll
        for (int j = 0; j < 8; ++j) c[j] = cinit;

        const float* cbcol = cb + (size_t)kcol * DDIM + dphase;
#pragma unroll
        for (int t = 0; t < 64; ++t) {
            v2f b = *(const v2f*)(cbcol + 4 * t);  // B 4x16 fp32 fragment
            c = __builtin_amdgcn_wmma_f32_16x16x4_f32(
                    false, za[t], false, b, (short)0, c, false, false);
        }
        // Running per-lane argmax (strict > keeps smallest k on ties)
#pragma unroll
        for (int j = 0; j < 8; ++j) {
            if (c[j] > bestv[j]) { bestv[j] = c[j]; besti[j] = kcol; }
        }
    }

    // Cross-lane reduce within each 16-lane half (C layout: VGPR j = row j / j+8)
#pragma unroll
    for (int j = 0; j < 8; ++j) {
#pragma unroll
        for (int off = 1; off < 16; off <<= 1) {
            float ov = __shfl_xor(bestv[j], off, 32);
            int   oi = __shfl_xor(besti[j], off, 32);
            if (ov > bestv[j] || (ov == bestv[j] && oi < besti[j])) {
                bestv[j] = ov; besti[j] = oi;
            }
        }
    }

    // Gather the 16 winning codebook rows to both output planes.
    float* out2 = out + (size_t)nrows * DDIM;
#pragma unroll
    for (int r = 0; r < 16; ++r) {
        int k = __shfl(besti[r & 7], (r < 8) ? 0 : 16, 32);
        const f4* src = (const f4*)(cb + (size_t)k * DDIM);
        f4 a0 = src[lane * 2];
        f4 a1 = src[lane * 2 + 1];
        size_t o = (size_t)(row0 + r) * DDIM;
        ((f4*)(out  + o))[lane * 2]     = a0;
        ((f4*)(out  + o))[lane * 2 + 1] = a1;
        ((f4*)(out2 + o))[lane * 2]     = a0;
        ((f4*)(out2 + o))[lane * 2 + 1] = a1;
    }
}

extern "C" void kernel_launch(void* const* d_in, const int* in_sizes, int n_in,
                              void* d_out, int out_size, void* d_ws, size_t ws_size,
                              hipStream_t stream) {
    const float* x  = (const float*)d_in[0];
    const float* cb = (const float*)d_in[1];
    float* out = (float*)d_out;
    float* normhalf = (float*)d_ws;            // 512 floats of scratch

    const int nrows = in_sizes[0] / DDIM;      // 262144

    hipLaunchKernelGGL(vq_norms_kernel, dim3((KCB + 255) / 256), dim3(256),
                       0, stream, cb, normhalf);

    const int nstrips = (nrows + 15) / 16;     // 16384
    const int nblocks = (nstrips + 7) / 8;     // 2048 blocks x 8 waves
    hipLaunchKernelGGL(vq_main_kernel, dim3(nblocks), dim3(256),
                       0, stream, x, cb, normhalf, out, nrows);
}